// Quantizer_85023172591985
// MI455X (gfx1250) — compile-verified
//
#include <hip/hip_runtime.h>
#include <hip/hip_bf16.h>

typedef __attribute__((ext_vector_type(2))) float v2f;
typedef __attribute__((ext_vector_type(8))) float v8f;

#define N_PTS       65536
#define DIM         256
#define KCODES      4096
#define NKT         (KCODES / 16)     // 256 code tiles of 16
#define ROWS_PER_WAVE  16
#define WAVES_PER_BLK  8
#define ROWS_PER_BLK   (ROWS_PER_WAVE * WAVES_PER_BLK)   // 128
#define LDS_STRIDE  260               // padded floats per codebook row (bank-conflict-free)

// ---------------- kernel 1: c_sq[k] = ||codebook[k]||^2 ----------------
__global__ __launch_bounds__(256) void vq_csq_kernel(const float* __restrict__ cb,
                                                     float* __restrict__ csq) {
    const int warp = (blockIdx.x * blockDim.x + threadIdx.x) >> 5;  // one wave32 per code row
    const int lane = threadIdx.x & 31;
    if (warp >= KCODES) return;
    const float4* r = (const float4*)(cb + (size_t)warp * DIM);
    float4 a = r[lane];
    float4 b = r[lane + 32];
    float s = a.x*a.x + a.y*a.y + a.z*a.z + a.w*a.w
            + b.x*b.x + b.y*b.y + b.z*b.z + b.w*b.w;
    #pragma unroll
    for (int m = 16; m >= 1; m >>= 1) s += __shfl_xor(s, m, 32);
    if (lane == 0) csq[warp] = s;
}

// ---------------- kernel 2: argmin + gather ----------------
__global__ __launch_bounds__(256, 1) void vq_argmin_kernel(
    const float* __restrict__ x, const float* __restrict__ cb,
    const float* __restrict__ csq, float* __restrict__ out_q,
    float* __restrict__ out_idx) {

    __shared__ __align__(16) float tile[2][16 * LDS_STRIDE];

    const int tid  = threadIdx.x;
    const int lane = tid & 31;
    const int wid  = tid >> 5;
    const int hi   = lane >> 4;        // half-wave select (k-pair / M+8)
    const int col  = lane & 15;        // matrix column / row-in-tile
    const int row_base = blockIdx.x * ROWS_PER_BLK + wid * ROWS_PER_WAVE;

    // ---- A fragments: 16 rows x 256 dims, resident in 128 VGPRs ----
    // WMMA f32 16x16x4 A layout: lanes 0-15 hold M=lane, k={4t,4t+1}; lanes 16-31 k={4t+2,4t+3}
    v2f a_frag[64];
    {
        const v2f* xp = (const v2f*)(x + (size_t)(row_base + col) * DIM);
        #pragma unroll
        for (int t = 0; t < 64; ++t) a_frag[t] = xp[2 * t + hi];
    }

    float best_val[8];
    int   best_idx[8];
    #pragma unroll
    for (int v = 0; v < 8; ++v) { best_val[v] = 3.4e38f; best_idx[v] = 0; }

    // stage one 16-code x 256-dim tile into LDS (async, 4 x b128 per thread)
    auto stage = [&](int kt, int buf) {
        #pragma unroll
        for (int j = 0; j < 4; ++j) {
            int f  = tid + 256 * j;               // 0..1023 float4s
            int r  = f >> 6;                      // code row in tile
            int c4 = f & 63;                      // float4 within row
            unsigned voff = (unsigned)(((kt * 16 + r) * DIM + c4 * 4) * 4);
            unsigned lds  = (unsigned)(uintptr_t)&tile[buf][r * LDS_STRIDE + c4 * 4];
            asm volatile("global_load_async_to_lds_b128 %0, %1, %2"
                         :: "v"(lds), "v"(voff), "s"(cb) : "memory");
        }
    };

    stage(0, 0);

    #pragma unroll 1
    for (int kt = 0; kt < NKT; ++kt) {
        const int cur = kt & 1;
        if (kt + 1 < NKT) {
            stage(kt + 1, cur ^ 1);
            asm volatile("s_wait_asynccnt 0x4" ::: "memory");  // tile kt done, next 4 in flight
        } else {
            asm volatile("s_wait_asynccnt 0x0" ::: "memory");
        }
        __syncthreads();

        // B fragment reads: lane -> code row `col` (stride 260 floats => conflict-free)
        const v2f* bp = (const v2f*)&tile[cur][col * LDS_STRIDE];
        v8f acc0 = {};
        v8f acc1 = {};
        #pragma unroll
        for (int t = 0; t < 64; t += 2) {
            v2f b0 = bp[2 * t + hi];
            acc0 = __builtin_amdgcn_wmma_f32_16x16x4_f32(
                       false, a_frag[t], false, b0, (short)0, acc0, false, false);
            v2f b1 = bp[2 * (t + 1) + hi];
            acc1 = __builtin_amdgcn_wmma_f32_16x16x4_f32(
                       false, a_frag[t + 1], false, b1, (short)0, acc1, false, false);
        }

        const int   code0 = kt * 16 + col;
        const float cs    = csq[code0];
        #pragma unroll
        for (int v = 0; v < 8; ++v) {
            float dot   = acc0[v] + acc1[v];
            float score = fmaf(-2.0f, dot, cs);   // ||c||^2 - 2 x.c  (||x||^2 drops out)
            if (score < best_val[v]) { best_val[v] = score; best_idx[v] = code0; }
        }
        __syncthreads();
    }

    // ---- reduce over the 16 columns (lanes) per row; tie -> lowest index ----
    float row_val[8];
    int   row_idx[8];
    #pragma unroll
    for (int v = 0; v < 8; ++v) {
        float val = best_val[v];
        int   idx = best_idx[v];
        #pragma unroll
        for (int m = 1; m < 16; m <<= 1) {
            float ov = __shfl_xor(val, m, 16);
            int   oi = __shfl_xor(idx, m, 16);
            if (ov < val || (ov == val && oi < idx)) { val = ov; idx = oi; }
        }
        row_val[v] = val;
        row_idx[v] = idx;
    }

    if (col == 0) {
        #pragma unroll
        for (int v = 0; v < 8; ++v)
            out_idx[row_base + 8 * hi + v] = (float)row_idx[v];
    }

    // ---- gather: whole wave copies codebook[idx] row per output row ----
    #pragma unroll 1
    for (int m = 0; m < 16; ++m) {
        int idx_m = __shfl(row_idx[m & 7], (m >> 3) * 16, 32);
        const float4* src = (const float4*)(cb + (size_t)idx_m * DIM);
        float4*       dst = (float4*)(out_q + (size_t)(row_base + m) * DIM);
        dst[lane]      = src[lane];
        dst[lane + 32] = src[lane + 32];
    }
}

extern "C" void kernel_launch(void* const* d_in, const int* in_sizes, int n_in,
                              void* d_out, int out_size, void* d_ws, size_t ws_size,
                              hipStream_t stream) {
    (void)in_sizes; (void)n_in; (void)out_size; (void)ws_size;
    const float* x   = (const float*)d_in[0];
    const float* cb  = (const float*)d_in[1];
    float* out       = (float*)d_out;
    float* out_q     = out;                              // N*DIM quantized rows
    float* out_idx   = out + (size_t)N_PTS * DIM;        // N indices (as float)
    float* csq       = (float*)d_ws;                     // K floats scratch

    vq_csq_kernel<<<KCODES / 8, 256, 0, stream>>>(cb, csq);
    vq_argmin_kernel<<<N_PTS / ROWS_PER_BLK, 256, 0, stream>>>(x, cb, csq, out_q, out_idx);
}